// GRUblock_53094385713157
// MI455X (gfx1250) — compile-verified
//
#include <hip/hip_runtime.h>
#include <hip/hip_bf16.h>
#include <math.h>

typedef float v2f __attribute__((ext_vector_type(2)));
typedef float v8f __attribute__((ext_vector_type(8)));

#define T_STEPS 512
#define DIM 512
#define ROWS (T_STEPS * 32)      // 16384 rows of xx (row r = t*32 + b)
#define NBLK_SCAN 16             // 2 groups x 8 WGs

// Padded LDS tile layout: 2 pad DWORDs every 256 DWORDs (matches TDM
// pad_interval=7, pad_amount=1). Row stride 516 floats -> banks (4*row+k)%64,
// conflict-free for 16-row fragment gathers; offsets stay 8B aligned.
#define LDSTRIDE 516
#define LOFF(c) ((c) + 2 * ((c) >> 8))

// ---- workspace layout (bytes) ----
static constexpr size_t OFF_P    = 0;                                        // 3 * 16384 * 512 f32
static constexpr size_t OFF_PART = OFF_P    + (size_t)3 * ROWS * DIM * 4;    // 3 * 16 * 512 * 2 f32
static constexpr size_t OFF_SCSH = OFF_PART + (size_t)3 * 16 * DIM * 2 * 4;  // 3 * 512 * 2 f32
static constexpr size_t OFF_HBUF = OFF_SCSH + (size_t)3 * DIM * 2 * 4;       // 2(parity)*2(group)*16*512 f32
static constexpr size_t OFF_RH   = OFF_HBUF + (size_t)2 * 2 * 16 * DIM * 4;  // 2(group)*16*512 f32
static constexpr size_t OFF_BAR  = OFF_RH   + (size_t)2 * 16 * DIM * 4;      // 2 u32

// ============================================================================
// Tensor Data Mover staging of a 16x512 f32 tile (row stride 512 elements in
// global) into padded LDS. D# built per CDNA5 ISA section 8. Probe-verified
// builtins; guarded so the plain-loop path is used if they are absent.
// ============================================================================
#if defined(__AMDGCN__) && __has_builtin(__builtin_amdgcn_tensor_load_to_lds) && \
    __has_builtin(__builtin_amdgcn_s_wait_tensorcnt)
#define USE_TDM 1
#endif

#ifdef USE_TDM
typedef unsigned u32x4 __attribute__((ext_vector_type(4)));
typedef int      i32x4 __attribute__((ext_vector_type(4)));
typedef int      i32x8 __attribute__((ext_vector_type(8)));

__device__ __forceinline__ unsigned lds_offset_of(const void* p) {
    return (unsigned)(unsigned long long)(__attribute__((address_space(3))) const void*)p;
}

__device__ __forceinline__ void tdm_load_tile_16x512(const float* gsrc, unsigned lds_byte_off) {
    unsigned long long ga = (unsigned long long)(size_t)gsrc;
    u32x4 g0;
    g0[0] = 1u;                                              // count=1, user mode
    g0[1] = lds_byte_off;                                    // lds_addr (bytes)
    g0[2] = (unsigned)(ga & 0xFFFFFFFFu);                    // global_addr[31:0]
    g0[3] = (unsigned)((ga >> 32) & 0x1FFFFFFu) | (2u << 30); // addr[56:32] | type=2
    i32x8 g1;
    g1[0] = (int)((2u << 16)      // data_size = 4B
                | (1u << 20)      // pad_enable
                | (7u << 22)      // pad_interval: 256 DWORDs
                | (1u << 25));    // pad_amount: 2 DWORDs
    g1[1] = (int)(512u << 16);    // tensor_dim0[15:0] = 512
    g1[2] = (int)(16u << 16);     // tensor_dim0[31:16]=0 | tensor_dim1[15:0]=16
    g1[3] = (int)(512u << 16);    // tensor_dim1[31:16]=0 | tile_dim0=512
    g1[4] = 16;                   // tile_dim1=16 | tile_dim2=0
    g1[5] = 512;                  // tensor_dim0_stride[31:0]
    g1[6] = 0;                    // stride0[47:32] | stride1[15:0]
    g1[7] = 0;                    // stride1[47:16]
    i32x4 gz = {0, 0, 0, 0};
#if __clang_major__ >= 23
    i32x8 gz8 = {0, 0, 0, 0, 0, 0, 0, 0};
    __builtin_amdgcn_tensor_load_to_lds(g0, g1, gz, gz, gz8, 0);
#else
    __builtin_amdgcn_tensor_load_to_lds(g0, g1, gz, gz, 0);
#endif
}
#endif  // USE_TDM

// ============================================================================
// init: zero h state (both parities) and barrier counters (ws is poisoned 0xAA)
// ============================================================================
__global__ __launch_bounds__(256) void scan_init(float* __restrict__ hbuf,
                                                 unsigned* __restrict__ bar) {
    int idx = blockIdx.x * 256 + threadIdx.x;
    if (idx < 2 * 2 * 16 * DIM) hbuf[idx] = 0.0f;
    if (idx < 2) bar[idx] = 0u;
}

// ============================================================================
// Projection GEMM: P[w][r][c] = sum_k xx[r][k] * W_w[c][k]
// Block = 8 waves, one N-tile: the shared 16x512 W tile is staged once into
// padded LDS (TDM if available), B fragments then come from the DS pipe.
// One wave = one 16x16 f32 tile via V_WMMA_F32_16X16X4_F32 (K-loop of 128).
// grid (32 Ntiles, 128 Mtile-octets, 3 weights), block 256.
// ============================================================================
__global__ __launch_bounds__(256) void proj_gemm(const float* __restrict__ x,
                                                 const float* __restrict__ Wh,
                                                 const float* __restrict__ Wz,
                                                 const float* __restrict__ Wr,
                                                 float* __restrict__ P) {
    __shared__ __align__(16) float Bs[16 * LDSTRIDE];

    const int lane  = threadIdx.x & 31;
    const int wave  = threadIdx.x >> 5;
    const int hi    = lane >> 4;
    const int mn    = lane & 15;
    const int ntile = blockIdx.x;
    const int mtile = blockIdx.y * 8 + wave;
    const int w     = blockIdx.z;
    const float* W  = (w == 0) ? Wh : (w == 1) ? Wz : Wr;
    const float* wt = W + (size_t)ntile * 16 * DIM;  // 16 rows x 512, row-major

#ifdef USE_TDM
    if (threadIdx.x < 32) {
        tdm_load_tile_16x512(wt, lds_offset_of(Bs));
        __builtin_amdgcn_s_wait_tensorcnt(0);
    }
#else
    for (int i = threadIdx.x; i < 16 * 256; i += 256) {
        int row = i >> 8, c = (i & 255) * 2;
        *(float2*)&Bs[row * LDSTRIDE + LOFF(c)] = *(const float2*)&wt[row * DIM + c];
    }
#endif
    __syncthreads();

    // xx row -> source row of x (forward b<16, reversed-time b>=16)
    const int r = mtile * 16 + mn;
    const int t = r >> 5, b = r & 31;
    const int xrow = (b < 16) ? (t * 16 + b) : ((T_STEPS - 1 - t) * 16 + (b - 16));
    const float* aptr = x + (size_t)xrow * DIM + 2 * hi;

    v8f acc = {0, 0, 0, 0, 0, 0, 0, 0};
#pragma unroll 8
    for (int k = 0; k < DIM; k += 4) {
        v2f av = *(const v2f*)(aptr + k);
        v2f bv = *(const v2f*)&Bs[mn * LDSTRIDE + LOFF(k) + 2 * hi];
        acc = __builtin_amdgcn_wmma_f32_16x16x4_f32(false, av, false, bv,
                                                    (short)0, acc, false, false);
    }
    float* out = P + (size_t)w * ROWS * DIM;
#pragma unroll
    for (int v = 0; v < 8; ++v) {
        int m = v + 8 * hi;
        out[(size_t)(mtile * 16 + m) * DIM + ntile * 16 + mn] = acc[v];
    }
}

// ============================================================================
// BatchNorm stats, stage 1: per (weight, 1024-row chunk, channel) partials.
// Coalesced across threads; software prefetch 16 rows ahead.
// ============================================================================
__global__ __launch_bounds__(256) void bn_partial(const float* __restrict__ P,
                                                  float* __restrict__ part) {
    const int ch = blockIdx.x * 256 + threadIdx.x;  // 0..511
    const int chunk = blockIdx.y;                   // 0..15
    const int w = blockIdx.z;                       // 0..2
    const float* p = P + (size_t)w * ROWS * DIM + (size_t)chunk * 1024 * DIM + ch;
    float s = 0.0f, ss = 0.0f;
    for (int r = 0; r < 1024; ++r) {
        __builtin_prefetch(p + (size_t)(r + 16) * DIM, 0, 1);
        float v = p[(size_t)r * DIM];
        s += v; ss += v * v;
    }
    float* o = part + ((size_t)(w * 16 + chunk) * DIM + ch) * 2;
    o[0] = s; o[1] = ss;
}

// BatchNorm stats, stage 2: fold to per-channel scale/shift (population var).
__global__ __launch_bounds__(256) void bn_finalize(const float* __restrict__ part,
                                                   const float* __restrict__ gh, const float* __restrict__ bh,
                                                   const float* __restrict__ gz, const float* __restrict__ bz,
                                                   const float* __restrict__ gr, const float* __restrict__ br,
                                                   float* __restrict__ scsh) {
    const int idx = blockIdx.x * 256 + threadIdx.x;  // 0..1535
    const int w = idx >> 9, ch = idx & 511;
    float s = 0.0f, ss = 0.0f;
    for (int c = 0; c < 16; ++c) {
        const float* o = part + ((size_t)(w * 16 + c) * DIM + ch) * 2;
        s += o[0]; ss += o[1];
    }
    const float N = (float)ROWS;
    float mu = s / N;
    float var = ss / N - mu * mu;
    const float* g  = (w == 0) ? gh : (w == 1) ? gz : gr;
    const float* bb = (w == 0) ? bh : (w == 1) ? bz : br;
    float sc = g[ch] * rsqrtf(var + 1e-5f);
    scsh[((size_t)w * DIM + ch) * 2]     = sc;
    scsh[((size_t)w * DIM + ch) * 2 + 1] = bb[ch] - mu * sc;
}

// ============================================================================
// Grid barrier over NBLK_SCAN blocks (generation counter).
// ============================================================================
__device__ __forceinline__ void grid_barrier(unsigned* bar) {
    __syncthreads();
    if (threadIdx.x == 0) {
        unsigned gen = __hip_atomic_load(&bar[1], __ATOMIC_ACQUIRE, __HIP_MEMORY_SCOPE_AGENT);
        unsigned prev = __hip_atomic_fetch_add(&bar[0], 1u, __ATOMIC_ACQ_REL, __HIP_MEMORY_SCOPE_AGENT);
        if (prev + 1 == NBLK_SCAN) {
            __hip_atomic_store(&bar[0], 0u, __ATOMIC_RELAXED, __HIP_MEMORY_SCOPE_AGENT);
            __hip_atomic_fetch_add(&bar[1], 1u, __ATOMIC_ACQ_REL, __HIP_MEMORY_SCOPE_AGENT);
        } else {
            while (__hip_atomic_load(&bar[1], __ATOMIC_ACQUIRE, __HIP_MEMORY_SCOPE_AGENT) == gen) {
                __builtin_amdgcn_s_sleep(1);
            }
        }
    }
    __syncthreads();
}

// ============================================================================
// Recurrent scan. 16 blocks x 128 threads (4 waves).
//   group = blockIdx/8 -> batch rows [group*16, group*16+16) (independent).
//   each wave owns one 16-col tile: cols = (blockIdx%8)*64 + wave*16 .. +15
// Per step: phase A (z, r via Uz/Ur WMMA on h_t; publish r.h) | barrier |
//           phase B (Uh WMMA on r.h; h update; write out)     | barrier.
// h_t and r.h slabs are TDM-staged into padded LDS each phase.
// ============================================================================
__global__ __launch_bounds__(128) void gru_scan(const float* __restrict__ P,
                                                const float* __restrict__ scsh,
                                                const float* __restrict__ Uh,
                                                const float* __restrict__ Uz,
                                                const float* __restrict__ Ur,
                                                const float* __restrict__ mask,
                                                float* __restrict__ hbuf,
                                                float* __restrict__ rhbuf,
                                                unsigned* __restrict__ bar,
                                                float* __restrict__ out) {
    __shared__ __align__(16) float hsA[16 * LDSTRIDE];  // h_t slab
    __shared__ __align__(16) float hsB[16 * LDSTRIDE];  // (r . h_t) slab

    const int lane  = threadIdx.x & 31;
    const int wave  = threadIdx.x >> 5;
    const int group = blockIdx.x >> 3;
    const int wg    = blockIdx.x & 7;
    const int hi    = lane >> 4;
    const int nn    = lane & 15;
    const int col   = wg * 64 + wave * 16 + nn;

    const float* Ph = P;
    const float* Pz = P + (size_t)1 * ROWS * DIM;
    const float* Pr = P + (size_t)2 * ROWS * DIM;
    const float scH = scsh[((size_t)0 * DIM + col) * 2], shH = scsh[((size_t)0 * DIM + col) * 2 + 1];
    const float scZ = scsh[((size_t)1 * DIM + col) * 2], shZ = scsh[((size_t)1 * DIM + col) * 2 + 1];
    const float scR = scsh[((size_t)2 * DIM + col) * 2], shR = scsh[((size_t)2 * DIM + col) * 2 + 1];

    const float* bzp = Uz + (size_t)col * DIM + 2 * hi;  // B[k][N] = U[col][k]
    const float* brp = Ur + (size_t)col * DIM + 2 * hi;
    const float* bhp = Uh + (size_t)col * DIM + 2 * hi;
    float* rhg = rhbuf + (size_t)group * 16 * DIM;

#ifdef USE_TDM
    const unsigned ldsA = lds_offset_of(hsA);
    const unsigned ldsB = lds_offset_of(hsB);
#endif

    for (int t = 0; t < T_STEPS; ++t) {
        // ---- stage h_t into padded LDS ----
        const float* hsrc = hbuf + (size_t)((t & 1) * 2 + group) * 16 * DIM;
#ifdef USE_TDM
        if (threadIdx.x < 32) {
            tdm_load_tile_16x512(hsrc, ldsA);
            __builtin_amdgcn_s_wait_tensorcnt(0);
        }
#else
        for (int i = threadIdx.x; i < 16 * 256; i += 128) {
            int row = i >> 8, c = (i & 255) * 2;
            *(float2*)&hsA[row * LDSTRIDE + LOFF(c)] = *(const float2*)&hsrc[row * DIM + c];
        }
#endif
        __syncthreads();

        // ---- phase A: z and r gates (two WMMA matvecs over h_t) ----
        v8f zacc = {0, 0, 0, 0, 0, 0, 0, 0};
        v8f racc = {0, 0, 0, 0, 0, 0, 0, 0};
#pragma unroll 4
        for (int k = 0; k < DIM; k += 4) {
            v2f av = *(const v2f*)&hsA[nn * LDSTRIDE + LOFF(k) + 2 * hi];
            v2f bz = *(const v2f*)(bzp + k);
            v2f br = *(const v2f*)(brp + k);
            zacc = __builtin_amdgcn_wmma_f32_16x16x4_f32(false, av, false, bz, (short)0, zacc, false, false);
            racc = __builtin_amdgcn_wmma_f32_16x16x4_f32(false, av, false, br, (short)0, racc, false, false);
        }
        float zreg[8];
#pragma unroll
        for (int v = 0; v < 8; ++v) {
            int m = v + 8 * hi;
            int b = group * 16 + m;
            size_t row = (size_t)(t * 32 + b) * DIM + col;
            float wz = Pz[row] * scZ + shZ;
            float wr = Pr[row] * scR + shR;
            float z = 1.0f / (1.0f + expf(-(wz + zacc[v])));
            float r = 1.0f / (1.0f + expf(-(wr + racc[v])));
            zreg[v] = z;
            rhg[(size_t)m * DIM + col] = r * hsA[m * LDSTRIDE + LOFF(col)];  // publish r.h
        }
        __threadfence();
        grid_barrier(bar);

        // ---- phase B: Uh matvec over (r.h), candidate, state update ----
#ifdef USE_TDM
        if (threadIdx.x < 32) {
            tdm_load_tile_16x512(rhg, ldsB);
            __builtin_amdgcn_s_wait_tensorcnt(0);
        }
#else
        for (int i = threadIdx.x; i < 16 * 256; i += 128) {
            int row = i >> 8, c = (i & 255) * 2;
            *(float2*)&hsB[row * LDSTRIDE + LOFF(c)] = *(const float2*)&rhg[row * DIM + c];
        }
#endif
        __syncthreads();

        v8f hacc = {0, 0, 0, 0, 0, 0, 0, 0};
#pragma unroll 4
        for (int k = 0; k < DIM; k += 4) {
            v2f av = *(const v2f*)&hsB[nn * LDSTRIDE + LOFF(k) + 2 * hi];
            v2f bh = *(const v2f*)(bhp + k);
            hacc = __builtin_amdgcn_wmma_f32_16x16x4_f32(false, av, false, bh, (short)0, hacc, false, false);
        }
        float* hdst = hbuf + (size_t)(((t + 1) & 1) * 2 + group) * 16 * DIM;
#pragma unroll
        for (int v = 0; v < 8; ++v) {
            int m = v + 8 * hi;
            int b = group * 16 + m;
            size_t row = (size_t)(t * 32 + b) * DIM + col;
            float wh = Ph[row] * scH + shH;
            float a  = wh + hacc[v];
            float hc = tanhf(a) * mask[(size_t)b * DIM + col];
            float hold = hsA[m * LDSTRIDE + LOFF(col)];
            float z = zreg[v];
            float hn = z * hold + (1.0f - z) * hc;
            hdst[(size_t)m * DIM + col] = hn;
            if (group == 0)
                out[((size_t)t * 16 + m) * 1024 + col] = hn;                       // forward half
            else
                out[((size_t)(T_STEPS - 1 - t) * 16 + m) * 1024 + 512 + col] = hn; // backward half
        }
        __threadfence();
        grid_barrier(bar);
    }
}

// ============================================================================
extern "C" void kernel_launch(void* const* d_in, const int* in_sizes, int n_in,
                              void* d_out, int out_size, void* d_ws, size_t ws_size,
                              hipStream_t stream) {
    const float* x    = (const float*)d_in[0];
    const float* Wh   = (const float*)d_in[1];
    const float* Wz   = (const float*)d_in[2];
    const float* Wr   = (const float*)d_in[3];
    const float* Uh   = (const float*)d_in[4];
    const float* Uz   = (const float*)d_in[5];
    const float* Ur   = (const float*)d_in[6];
    const float* gh   = (const float*)d_in[7];
    const float* bh   = (const float*)d_in[8];
    const float* gz   = (const float*)d_in[9];
    const float* bz   = (const float*)d_in[10];
    const float* gr   = (const float*)d_in[11];
    const float* br   = (const float*)d_in[12];
    const float* mask = (const float*)d_in[13];

    char* ws = (char*)d_ws;
    float*    P    = (float*)(ws + OFF_P);
    float*    part = (float*)(ws + OFF_PART);
    float*    scsh = (float*)(ws + OFF_SCSH);
    float*    hbuf = (float*)(ws + OFF_HBUF);
    float*    rh   = (float*)(ws + OFF_RH);
    unsigned* bar  = (unsigned*)(ws + OFF_BAR);
    float*    out  = (float*)d_out;

    scan_init<<<128, 256, 0, stream>>>(hbuf, bar);
    proj_gemm<<<dim3(32, 128, 3), 256, 0, stream>>>(x, Wh, Wz, Wr, P);
    bn_partial<<<dim3(2, 16, 3), 256, 0, stream>>>(P, part);
    bn_finalize<<<6, 256, 0, stream>>>(part, gh, bh, gz, bz, gr, br, scsh);
    gru_scan<<<NBLK_SCAN, 128, 0, stream>>>(P, scsh, Uh, Uz, Ur, mask, hbuf, rh, bar, out);
}